// FarthestPointSample_63256278335597
// MI455X (gfx1250) — compile-verified
//
#include <hip/hip_runtime.h>
#include <stdint.h>

// Problem constants (from reference): pcs [32, 3, 131072] fp32 -> idx [32, 1024] int64
#define BATCHES    32
#define NPTS       131072
#define NCENT      1024

// Decomposition: 16 workgroups cooperate on one batch.
#define WGS_PER_B  16
#define PW         (NPTS / WGS_PER_B)   // 8192 points per WG
#define NTHREADS   512                  // 16 wave32 waves
#define KPT        (PW / NTHREADS)      // 16 points per thread (min-dist kept in VGPRs)

// Workspace layout (zeroed via hipMemsetAsync each launch):
//   [0, 16384)      : per-group arrival counters, 256B stride each
//   [16384, 24576)  : u64 slots[32 groups][2 parities][16 WGs]
#define CNT_STRIDE 64          // in u32 units => 256 bytes
#define SLOT_OFF   16384

__device__ __forceinline__ void async_copy_b128(uint32_t lds_addr, const float* gptr) {
  // CDNA5: GLOBAL_LOAD_ASYNC_TO_LDS_B128, GV mode (64-bit vaddr, no saddr).
  asm volatile("global_load_async_to_lds_b128 %0, %1, off"
               :: "v"(lds_addr), "v"((unsigned long long)(uintptr_t)gptr)
               : "memory");
}

__global__ __launch_bounds__(NTHREADS)
void fps_kernel(const float* __restrict__ pcs,
                long long*   __restrict__ out,
                unsigned int* __restrict__ counters,
                unsigned long long* __restrict__ slots)
{
  __shared__ float xs[PW];
  __shared__ float ys[PW];
  __shared__ float zs[PW];
  __shared__ unsigned long long red;
  __shared__ unsigned int s_widx;
  __shared__ float s_px, s_py, s_pz;

  const int tid = threadIdx.x;
  const int b   = blockIdx.x / WGS_PER_B;   // batch
  const int wgl = blockIdx.x % WGS_PER_B;   // workgroup-in-group
  const long long bb = (long long)b * 3 * NPTS;

  const float* gx = pcs + bb + 0LL * NPTS + (long long)wgl * PW;
  const float* gy = pcs + bb + 1LL * NPTS + (long long)wgl * PW;
  const float* gz = pcs + bb + 2LL * NPTS + (long long)wgl * PW;

  // ---- Stage this WG's point slice into LDS via async copies (B128 chunks) ----
  {
    uint32_t lx = (uint32_t)(uintptr_t)(void*)&xs[0];
    uint32_t ly = (uint32_t)(uintptr_t)(void*)&ys[0];
    uint32_t lz = (uint32_t)(uintptr_t)(void*)&zs[0];
#pragma unroll
    for (int j = 0; j < PW / 4 / NTHREADS; ++j) {   // 4 iterations
      int c = tid + j * NTHREADS;                   // 16-byte chunk index
      async_copy_b128(lx + (uint32_t)c * 16u, gx + c * 4);
      async_copy_b128(ly + (uint32_t)c * 16u, gy + c * 4);
      async_copy_b128(lz + (uint32_t)c * 16u, gz + c * 4);
    }
  }
  if (tid == 0) red = 0ull;
  asm volatile("s_wait_asynccnt 0x0" ::: "memory");
  __syncthreads();

  // ---- min-dist lives in registers ----
  float md[KPT];
#pragma unroll
  for (int k = 0; k < KPT; ++k) md[k] = __builtin_inff();

  // First centroid is point 0; its coords are uniform scalar loads.
  float px = pcs[bb];
  float py = pcs[bb + NPTS];
  float pz = pcs[bb + 2LL * NPTS];

  if (wgl == 0 && tid == 0) out[(long long)b * NCENT] = 0ll;

  unsigned int*       cnt      = counters + b * CNT_STRIDE;
  unsigned long long* slotbase = slots + (size_t)b * 2 * WGS_PER_B;

  for (int r = 1; r < NCENT; ++r) {
    // -------- distance update + local argmax (first-index tie-break) --------
    float bv = -1.0f;
    unsigned int bi = 0;
#pragma unroll
    for (int k = 0; k < KPT; ++k) {
      int i = tid + k * NTHREADS;                // conflict-free LDS access
      float dx = xs[i] - px;
      float dy = ys[i] - py;
      float dz = zs[i] - pz;
      float d  = dx * dx + dy * dy + dz * dz;
      float m  = fminf(md[k], d);
      md[k] = m;
      if (m > bv) { bv = m; bi = (unsigned)(wgl * PW + i); }  // strict > keeps first
    }
    // Pack so that u64-max == (max dist, then min index). Distances >= 0, so
    // their IEEE bits order monotonically as u32.
    unsigned long long key =
        ((unsigned long long)__float_as_uint(bv) << 32) |
        (unsigned long long)(0xFFFFFFFFu - bi);

    // wave32 shuffle reduction, then one LDS u64 atomic-max per wave
#pragma unroll
    for (int off = 16; off > 0; off >>= 1) {
      unsigned long long o = __shfl_xor(key, off, 32);
      if (o > key) key = o;
    }
    if ((tid & 31) == 0) atomicMax(&red, key);
    __syncthreads();

    // -------- cross-WG argmax for this batch (leader only) --------
    if (tid == 0) {
      unsigned long long kk = red;
      red = 0ull;
      const int pr = r & 1;  // double-buffered slots: max skew is one round
      __hip_atomic_store(&slotbase[pr * WGS_PER_B + wgl], kk,
                         __ATOMIC_RELEASE, __HIP_MEMORY_SCOPE_AGENT);
      __hip_atomic_fetch_add(cnt, 1u, __ATOMIC_ACQ_REL, __HIP_MEMORY_SCOPE_AGENT);
      const unsigned int tgt = (unsigned)(WGS_PER_B * r);
      while (__hip_atomic_load(cnt, __ATOMIC_ACQUIRE,
                               __HIP_MEMORY_SCOPE_AGENT) < tgt)
        __builtin_amdgcn_s_sleep(1);

      unsigned long long w = 0ull;
      for (int q = 0; q < WGS_PER_B; ++q) {
        unsigned long long sv =
            __hip_atomic_load(&slotbase[pr * WGS_PER_B + q],
                              __ATOMIC_ACQUIRE, __HIP_MEMORY_SCOPE_AGENT);
        if (sv > w) w = sv;
      }
      const unsigned int widx = 0xFFFFFFFFu - (unsigned int)w;
      s_widx = widx;
      s_px = pcs[bb + widx];
      s_py = pcs[bb + NPTS + widx];
      s_pz = pcs[bb + 2LL * NPTS + widx];
      if (wgl == 0) out[(long long)b * NCENT + r] = (long long)widx;
    }
    __syncthreads();
    px = s_px; py = s_py; pz = s_pz;
    (void)s_widx;
  }
}

extern "C" void kernel_launch(void* const* d_in, const int* in_sizes, int n_in,
                              void* d_out, int out_size, void* d_ws, size_t ws_size,
                              hipStream_t stream) {
  (void)in_sizes; (void)n_in; (void)out_size; (void)ws_size;
  const float* pcs = (const float*)d_in[0];
  long long*   out = (long long*)d_out;

  unsigned int*       counters = (unsigned int*)d_ws;
  unsigned long long* slots    = (unsigned long long*)((char*)d_ws + SLOT_OFF);

  // Zero counters + slots every launch (graph-capturable, deterministic).
  hipMemsetAsync(d_ws, 0, 32768, stream);

  dim3 grid(BATCHES * WGS_PER_B);
  dim3 block(NTHREADS);
  hipLaunchKernelGGL(fps_kernel, grid, block, 0, stream, pcs, out, counters, slots);
}